// GGIN_77532749627917
// MI455X (gfx1250) — compile-verified
//
#include <hip/hip_runtime.h>

// GIN GNN forward for MI455X (gfx1250, wave32).
// Bottleneck: E=1.6M-edge scatter-add (~5 GB HBM traffic over 3 layers ->
// ~215us at 23.3 TB/s). GEMMs (N x 128 x 128, fp32 via V_WMMA_F32_16X16X4_F32)
// hide under that, so full fp32 precision is free.
// Algorithmic win: ctx @ Wc[l] computed at graph granularity (1024 rows)
// instead of node granularity (100000 rows).

#define D 128

typedef float v2f __attribute__((ext_vector_type(2)));
typedef float v8f __attribute__((ext_vector_type(8)));

__device__ __forceinline__ void atomic_add_f32(float* p, float v) {
  // lowers to global_atomic_add_f32 on gfx1250
  unsafeAtomicAdd(p, v);
}

// ---- h_lead = sum(lead_feat, axis=0) : 1 block of 128 threads -------------
__global__ void lead_sum_kernel(const float* __restrict__ lead,
                                float* __restrict__ out, int NL) {
  int d = threadIdx.x;  // 0..127
  float s = 0.0f;
  for (int i = 0; i < NL; ++i) s += lead[i * D + d];
  out[d] = s;
}

// ---- ctx_g[g,d] += h_lead[d] ----------------------------------------------
__global__ void add_lead_kernel(float* __restrict__ ctx,
                                const float* __restrict__ leadsum, int n) {
  int i = blockIdx.x * blockDim.x + threadIdx.x;
  if (i < n) ctx[i] += leadsum[i & (D - 1)];
}

// ---- out[seg[row]] += feat[row]  (32 lanes * float4 per row) --------------
__global__ void seg_sum_kernel(const float* __restrict__ feat,
                               const int* __restrict__ seg,
                               float* __restrict__ out, int rows) {
  long tid = (long)blockIdx.x * blockDim.x + threadIdx.x;
  if (tid >= (long)rows * 32) return;
  int row = (int)(tid >> 5);
  int j = ((int)tid & 31) * 4;
  float4 v = *(const float4*)(feat + (long)row * D + j);
  float* o = out + (long)seg[row] * D + j;
  atomic_add_f32(o + 0, v.x);
  atomic_add_f32(o + 1, v.y);
  atomic_add_f32(o + 2, v.z);
  atomic_add_f32(o + 3, v.w);
}

// ---- agg[dst[e]] += h[src[e]]  (32 lanes * float4 per edge) ---------------
__global__ void edge_agg_kernel(const float* __restrict__ h,
                                const int* __restrict__ src,
                                const int* __restrict__ dst,
                                float* __restrict__ agg, int nE) {
  long tid = (long)blockIdx.x * blockDim.x + threadIdx.x;
  if (tid >= (long)nE * 32) return;
  int e = (int)(tid >> 5);
  int j = ((int)tid & 31) * 4;
  int s = src[e];
  int t = dst[e];
  float4 v = *(const float4*)(h + (long)s * D + j);
  float* o = agg + (long)t * D + j;
  atomic_add_f32(o + 0, v.x);
  atomic_add_f32(o + 1, v.y);
  atomic_add_f32(o + 2, v.z);
  atomic_add_f32(o + 3, v.w);
}

// ---- WMMA fp32 GEMM: out[M,128] = act(A' @ W + bias), K = N = 128 ---------
// FUSED: A' = (1+eps)*h + agg + hg[gidA[row]]; else A' = A.
// Block: 256 threads = 8 waves; each wave computes 32 rows x 128 cols
// (two 16x4 A fragments share each 4x16 B fragment -> 16 WMMA per 8 LDS b64
// loads). Block covers 256 rows.
// W is stored transposed in LDS with an XOR swizzle: element (k,col) lives at
// col*128 + (k ^ (4*(col&15))). Since col&15 == l15 for every column tile,
// the XOR operand is one per-lane constant: per k-step address math is a
// single v_xor + v_lshl_add, and the 8 tile loads use immediate DS offsets
// (t*8192B < 64KB). Banks: half0 hits banks ≡{0,1} mod 4, half1 ≡{2,3} mod 4,
// all 64 dwords of the wave distinct -> conflict-free ds_load_b64.
// A-matrix 16x4 f32 layout per ISA 7.12.2: lanes 0-15 hold K={k0,k0+1},
// lanes 16-31 hold K={k0+2,k0+3}. C/D: VGPR v -> row v + 8*half, col = l15.
template <bool FUSED>
__global__ void __launch_bounds__(256)
gemm128_kernel(const float* __restrict__ A,
               const float* __restrict__ h, const float* __restrict__ agg,
               const float* __restrict__ hg, const int* __restrict__ gidA,
               const float* __restrict__ eps, int layer,
               const float* __restrict__ W,
               const float* __restrict__ biasVec,
               const float* __restrict__ biasRows, const int* __restrict__ gidB,
               float* __restrict__ out, int M, int doRelu) {
  __shared__ float Wlds[D * D];  // 64 KB of the 320 KB WGP LDS
  for (int i = threadIdx.x; i < D * D; i += 256) {
    const int k = i >> 7;
    const int col = i & 127;
    Wlds[col * D + (k ^ ((col & 15) << 2))] = W[i];
  }
  __syncthreads();

  const int wave = threadIdx.x >> 5;
  const int lane = threadIdx.x & 31;
  const int half = lane >> 4;   // 0: K pair {k0,k0+1}; 1: {k0+2,k0+3}
  const int l15  = lane & 15;
  const int c4   = l15 << 2;    // per-lane XOR swizzle constant
  const float* wbase = &Wlds[l15 * D];  // tile t adds t*16*D floats
  const int r0   = blockIdx.x * 256 + wave * 32;
  const int rowA = r0 + l15;         // A fragment 0 rows
  const int rowB = r0 + 16 + l15;    // A fragment 1 rows
  const int rcA  = rowA < M ? rowA : (M - 1);
  const int rcB  = rowB < M ? rowB : (M - 1);

  float ep1 = 1.0f;
  const float* hgA = nullptr;
  const float* hgB = nullptr;
  if (FUSED) {
    ep1 = 1.0f + eps[layer];
    hgA = hg + (long)gidA[rcA] * D;
    hgB = hg + (long)gidA[rcB] * D;
  }

  v8f acc0[8], acc1[8];
#pragma unroll
  for (int t = 0; t < 8; ++t) { acc0[t] = v8f{}; acc1[t] = v8f{}; }

  for (int k0 = 0; k0 < D; k0 += 4) {
    const int ka = k0 + 2 * half;
    v2f a0, a1;
    if (FUSED) {
      const long bA = (long)rcA * D + ka;
      const long bB = (long)rcB * D + ka;
      const v2f hA = *(const v2f*)(h + bA);
      const v2f aA = *(const v2f*)(agg + bA);
      const v2f gA = *(const v2f*)(hgA + ka);
      const v2f hB = *(const v2f*)(h + bB);
      const v2f aB = *(const v2f*)(agg + bB);
      const v2f gB = *(const v2f*)(hgB + ka);
      a0 = ep1 * hA + aA + gA;
      a1 = ep1 * hB + aB + gB;
    } else {
      a0 = *(const v2f*)(A + (long)rcA * D + ka);
      a1 = *(const v2f*)(A + (long)rcB * D + ka);
    }
    const int rot = ka ^ c4;  // single per-k-step swizzle term
#pragma unroll
    for (int t = 0; t < 8; ++t) {
      const v2f b = *(const v2f*)&wbase[t * 16 * D + rot];
      acc0[t] = __builtin_amdgcn_wmma_f32_16x16x4_f32(
          false, a0, false, b, (short)0, acc0[t], false, false);
      acc1[t] = __builtin_amdgcn_wmma_f32_16x16x4_f32(
          false, a1, false, b, (short)0, acc1[t], false, false);
    }
  }

  // Epilogue. C/D layout: VGPR v -> row base + v + 8*half; col = tile*16+l15.
#pragma unroll
  for (int frag = 0; frag < 2; ++frag) {
    const v8f* acc = frag ? acc1 : acc0;
    const int rbase = r0 + frag * 16;
#pragma unroll
    for (int v = 0; v < 8; ++v) {
      const int orow = rbase + v + 8 * half;
      if (orow >= M) continue;
      const float* brow =
          biasRows ? (biasRows + (long)gidB[orow] * D) : nullptr;
#pragma unroll
      for (int t = 0; t < 8; ++t) {
        const int col = t * 16 + l15;
        float x = acc[t][v];
        if (biasVec) x += biasVec[col];
        if (brow) x += brow[col];
        if (doRelu) x = fmaxf(x, 0.0f);
        out[(long)orow * D + col] = x;
      }
    }
  }
}

static inline unsigned cdiv(long a, long b) { return (unsigned)((a + b - 1) / b); }

extern "C" void kernel_launch(void* const* d_in, const int* in_sizes, int n_in,
                              void* d_out, int out_size, void* d_ws, size_t ws_size,
                              hipStream_t stream) {
  const float* x         = (const float*)d_in[0];
  const int*   src       = (const int*)d_in[1];
  const int*   dst       = (const int*)d_in[2];
  const int*   graph_ids = (const int*)d_in[3];
  const float* init_feat = (const float*)d_in[4];
  const int*   init_gids = (const int*)d_in[5];
  const float* lead_feat = (const float*)d_in[6];
  const float* W         = (const float*)d_in[7];
  const float* Wc        = (const float*)d_in[8];
  const float* b         = (const float*)d_in[9];
  const float* eps       = (const float*)d_in[10];
  const float* fc1_w     = (const float*)d_in[11];
  const float* fc1_b     = (const float*)d_in[12];
  const float* fc2_w     = (const float*)d_in[13];
  const float* fc2_b     = (const float*)d_in[14];

  const int N  = in_sizes[0] / D;
  const int E  = in_sizes[1];
  const int NI = in_sizes[4] / D;
  const int NL = in_sizes[6] / D;
  const int L  = in_sizes[10];
  const int C  = D;                 // fc2 output width == 128
  const int G  = out_size / C;

  // Workspace layout (floats): 2*N*D ping-pong + 4*G*D small + 128.
  float* ws      = (float*)d_ws;
  float* bufA    = ws;                         // N*D  (h / agg ping)
  float* bufB    = bufA + (size_t)N * D;       // N*D  (h / agg pong)
  float* ctxg    = bufB + (size_t)N * D;       // G*D
  float* hglob   = ctxg + (size_t)G * D;       // G*D
  float* cg      = hglob + (size_t)G * D;      // G*D
  float* hg1     = cg + (size_t)G * D;         // G*D
  float* leadsum = hg1 + (size_t)G * D;        // D

  // --- global context at graph granularity: ctx_g = seg_sum(init) + h_lead
  hipMemsetAsync(ctxg, 0, (size_t)G * D * sizeof(float), stream);
  lead_sum_kernel<<<1, D, 0, stream>>>(lead_feat, leadsum, NL);
  seg_sum_kernel<<<cdiv((long)NI * 32, 256), 256, 0, stream>>>(
      init_feat, init_gids, ctxg, NI);
  add_lead_kernel<<<cdiv((long)G * D, 256), 256, 0, stream>>>(ctxg, leadsum, G * D);

  const float* hcur = x;
  float* pingpong[2] = {bufA, bufB};
  for (int l = 0; l < L; ++l) {
    float* aggbuf = pingpong[l & 1];  // also becomes the layer output

    // h_global = seg_sum(h, graph_ids)
    hipMemsetAsync(hglob, 0, (size_t)G * D * sizeof(float), stream);
    seg_sum_kernel<<<cdiv((long)N * 32, 256), 256, 0, stream>>>(
        hcur, graph_ids, hglob, N);

    // agg = scatter-add over edges
    hipMemsetAsync(aggbuf, 0, (size_t)N * D * sizeof(float), stream);
    edge_agg_kernel<<<cdiv((long)E * 32, 256), 256, 0, stream>>>(
        hcur, src, dst, aggbuf, E);

    // cg = ctx_g @ Wc[l] + b[l]   (graph-level, tiny GEMM)
    gemm128_kernel<false><<<cdiv(G, 256), 256, 0, stream>>>(
        ctxg, nullptr, nullptr, nullptr, nullptr, nullptr, 0,
        Wc + (size_t)l * D * D, b + (size_t)l * D, nullptr, nullptr,
        cg, G, /*relu=*/0);

    // h = relu( ((1+eps)h + agg + h_global[g]) @ W[l] + cg[g] )
    // out aliases aggbuf: each wave reads only its own 32 rows, then writes them.
    gemm128_kernel<true><<<cdiv(N, 256), 256, 0, stream>>>(
        nullptr, hcur, aggbuf, hglob, graph_ids, eps, l,
        W + (size_t)l * D * D, nullptr, cg, graph_ids,
        aggbuf, N, /*relu=*/1);

    hcur = aggbuf;
  }

  // readout: out = relu(h_global @ fc1_w + fc1_b) @ fc2_w + fc2_b
  hipMemsetAsync(hglob, 0, (size_t)G * D * sizeof(float), stream);
  seg_sum_kernel<<<cdiv((long)N * 32, 256), 256, 0, stream>>>(
      hcur, graph_ids, hglob, N);
  gemm128_kernel<false><<<cdiv(G, 256), 256, 0, stream>>>(
      hglob, nullptr, nullptr, nullptr, nullptr, nullptr, 0,
      fc1_w, fc1_b, nullptr, nullptr, hg1, G, /*relu=*/1);
  gemm128_kernel<false><<<cdiv(G, 256), 256, 0, stream>>>(
      hg1, nullptr, nullptr, nullptr, nullptr, nullptr, 0,
      fc2_w, fc2_b, nullptr, nullptr, (float*)d_out, G, /*relu=*/0);
}